// FloydNet_49143015801466
// MI455X (gfx1250) — compile-verified
//
#include <hip/hip_runtime.h>
#include <hip/hip_bf16.h>

typedef float v2f __attribute__((ext_vector_type(2)));
typedef float v8f __attribute__((ext_vector_type(8)));

#define BB 4
#define NN 96
#define CC 256
#define HH 8
#define DD 32
#define DEPTH 4
#define FIVEC (5*CC)
#define MROWS (BB*NN*NN)        // 36864
#define NTOT  (MROWS*CC)        // 9437184
#define REDBLK 512
#define MSEG  36                // 9216 rows / 256
#define PAD   4                 // +4 floats keeps rows 16B-aligned; stride 36/100
                                // words -> 16 distinct banks for column reads

__device__ inline v8f wmma4(v2f a, v2f b, v8f c) {
  // V_WMMA_F32_16X16X4_F32 : D = A(16x4) * B(4x16) + C(16x16)
  return __builtin_amdgcn_wmma_f32_16x16x4_f32(false, a, false, b, (short)0, c,
                                               false, false);
}

// ---------------- embed: x[b,i,j,:] = emb_w[adj[b,i,j], :] ----------------
__global__ __launch_bounds__(256) void k_embed(const int* __restrict__ adj,
                                               const float* __restrict__ emb,
                                               float* __restrict__ x) {
  int sub = threadIdx.x >> 6, t4 = (threadIdx.x & 63) * 4;
  int row = blockIdx.x * 4 + sub;
  int a = adj[row];
  *(float4*)&x[(long)row * CC + t4] = *(const float4*)&emb[(long)a * CC + t4];
}

// ---------------- global mean/std reduction (deterministic 2-stage) -------
__global__ __launch_bounds__(256) void k_red1(const float* __restrict__ x,
                                              float* __restrict__ part) {
  const float4* x4 = (const float4*)x;
  float s = 0.f, q = 0.f;
  for (int i = blockIdx.x * 256 + threadIdx.x; i < NTOT / 4; i += REDBLK * 256) {
    float4 v = x4[i];
    s += v.x + v.y + v.z + v.w;
    q += v.x * v.x + v.y * v.y + v.z * v.z + v.w * v.w;
  }
  __shared__ float ls[256], lq[256];
  int tid = threadIdx.x;
  ls[tid] = s; lq[tid] = q;
  __syncthreads();
  for (int off = 128; off > 0; off >>= 1) {
    if (tid < off) { ls[tid] += ls[tid + off]; lq[tid] += lq[tid + off]; }
    __syncthreads();
  }
  if (tid == 0) { part[blockIdx.x * 2] = ls[0]; part[blockIdx.x * 2 + 1] = lq[0]; }
}

__global__ __launch_bounds__(256) void k_red2(const float* __restrict__ part,
                                              float* __restrict__ stats) {
  float s = 0.f, q = 0.f;
  int tid = threadIdx.x;
  for (int t = tid; t < REDBLK; t += 256) { s += part[2 * t]; q += part[2 * t + 1]; }
  __shared__ float ls[256], lq[256];
  ls[tid] = s; lq[tid] = q;
  __syncthreads();
  for (int off = 128; off > 0; off >>= 1) {
    if (tid < off) { ls[tid] += ls[tid + off]; lq[tid] += lq[tid + off]; }
    __syncthreads();
  }
  if (tid == 0) {
    float S = ls[0], Q = lq[0];
    float mean = S / (float)NTOT;
    float var = (Q - S * S / (float)NTOT) / (float)(NTOT - 1);  // ddof=1
    stats[0] = mean;
    stats[1] = rsqrtf(var);
  }
}

// ---------------- QKV GEMM: qkv[M,1280] = norm(x)[M,256] @ W^T ------------
// Block tile 128x64, 8 waves, each wave a 2x2 group of 16x16 tiles.
// A and B both stored K-contiguous so every WMMA operand pair is one
// adjacent 64-bit LDS read (no repacking moves).
#define BM 128
#define BN 64
#define BK 32
__global__ __launch_bounds__(256) void k_qkv(const float* __restrict__ x,
                                             const float* __restrict__ w,
                                             const float* __restrict__ stats,
                                             float* __restrict__ qkv) {
  __shared__ __align__(16) float As[BM][BK + PAD];
  __shared__ __align__(16) float Bs[BN][BK + PAD];
  const float mean = stats[0], rstd = stats[1];
  const int m0 = blockIdx.y * BM;
  const int n0 = blockIdx.x * BN;
  const int tid = threadIdx.x, lane = tid & 31, wv = tid >> 5;
  const int tm0 = (wv >> 1) * 2;        // tile-row base (of 8)
  const int tn0 = (wv & 1) * 2;         // tile-col base (of 4)
  const int row16 = lane & 15, half2 = (lane >> 4) << 1, half8 = (lane >> 4) * 8;
  v8f acc00 = {}, acc01 = {}, acc10 = {}, acc11 = {};
  for (int k0 = 0; k0 < CC; k0 += BK) {
#pragma unroll
    for (int r = 0; r < 4; ++r) {                    // A: 128x32 = 1024 float4
      int idx4 = tid + r * 256;
      int m = idx4 >> 3, kq = (idx4 & 7) * 4;
      float4 v = *(const float4*)&x[(long)(m0 + m) * CC + k0 + kq];
      v.x = (v.x - mean) * rstd; v.y = (v.y - mean) * rstd;
      v.z = (v.z - mean) * rstd; v.w = (v.w - mean) * rstd;
      *(float4*)&As[m][kq] = v;
    }
#pragma unroll
    for (int r = 0; r < 2; ++r) {                    // B: 64x32 = 512 float4
      int idx4 = tid + r * 256;
      int n = idx4 >> 3, kq = (idx4 & 7) * 4;
      *(float4*)&Bs[n][kq] = *(const float4*)&w[(long)(n0 + n) * CC + k0 + kq];
    }
    if (k0 + BK < CC) {                              // prefetch next K-chunk
      int mp = tid >> 5, kp = tid & 31;
      __builtin_prefetch((const void*)&x[(long)(m0 + mp) * CC + k0 + BK + kp], 0, 3);
      __builtin_prefetch((const void*)&w[(long)(n0 + mp) * CC + k0 + BK + kp], 0, 3);
    }
    __syncthreads();
#pragma unroll
    for (int kk = 0; kk < BK; kk += 4) {
      v2f a0, a1, b0, b1;
      a0.x = As[tm0 * 16 + row16][kk + half2];        a0.y = As[tm0 * 16 + row16][kk + half2 + 1];
      a1.x = As[(tm0 + 1) * 16 + row16][kk + half2];  a1.y = As[(tm0 + 1) * 16 + row16][kk + half2 + 1];
      b0.x = Bs[tn0 * 16 + row16][kk + half2];        b0.y = Bs[tn0 * 16 + row16][kk + half2 + 1];
      b1.x = Bs[(tn0 + 1) * 16 + row16][kk + half2];  b1.y = Bs[(tn0 + 1) * 16 + row16][kk + half2 + 1];
      acc00 = wmma4(a0, b0, acc00);
      acc01 = wmma4(a0, b1, acc01);
      acc10 = wmma4(a1, b0, acc10);
      acc11 = wmma4(a1, b1, acc11);
    }
    __syncthreads();
  }
#pragma unroll
  for (int r = 0; r < 8; ++r) {
    int col = lane & 15;
    qkv[(long)(m0 + tm0 * 16 + r + half8) * FIVEC + n0 + tn0 * 16 + col]             = acc00[r];
    qkv[(long)(m0 + tm0 * 16 + r + half8) * FIVEC + n0 + (tn0 + 1) * 16 + col]       = acc01[r];
    qkv[(long)(m0 + (tm0 + 1) * 16 + r + half8) * FIVEC + n0 + tn0 * 16 + col]       = acc10[r];
    qkv[(long)(m0 + (tm0 + 1) * 16 + r + half8) * FIVEC + n0 + (tn0 + 1) * 16 + col] = acc11[r];
  }
}

// ---------------- S1 per (b,h,i): s[j,p] = Q_i . K1_i^T -------------------
__global__ __launch_bounds__(256) void k_s1(const float* __restrict__ qkv,
                                            float* __restrict__ s) {
  int blk = blockIdx.x;
  int i = blk % NN, bh = blk / NN, h = bh % HH, b = bh / HH;
  __shared__ __align__(16) float Q[NN][DD + PAD];
  __shared__ __align__(16) float K1[NN][DD + PAD];
  const float* base = qkv + (long)(b * NN + i) * NN * FIVEC;
  int tid = threadIdx.x;
#pragma unroll
  for (int r = 0; r < 3; ++r) {                      // 96*32/4 = 768 float4
    int idx4 = tid + r * 256;
    int j = idx4 >> 3, dq = (idx4 & 7) * 4;
    *(float4*)&Q[j][dq]  = *(const float4*)&base[(long)j * FIVEC + h * DD + dq];
    *(float4*)&K1[j][dq] = *(const float4*)&base[(long)j * FIVEC + CC + h * DD + dq];
  }
  __syncthreads();
  int lane = tid & 31, wv = tid >> 5;
  int row16 = lane & 15, half2 = (lane >> 4) << 1, half8 = (lane >> 4) * 8, col = lane & 15;
  float* sout = s + ((long)bh * NN + i) * NN * NN;
  for (int g = wv; g < 9; g += 8) {                  // 3x3 groups of 2x2 tiles
    int tj0 = (g / 3) * 32, tp0 = (g % 3) * 32;
    v8f a00 = {}, a01 = {}, a10 = {}, a11 = {};
#pragma unroll
    for (int kk = 0; kk < DD; kk += 4) {
      v2f q0, q1, k0, k1;
      q0.x = Q[tj0 + row16][kk + half2];       q0.y = Q[tj0 + row16][kk + half2 + 1];
      q1.x = Q[tj0 + 16 + row16][kk + half2];  q1.y = Q[tj0 + 16 + row16][kk + half2 + 1];
      k0.x = K1[tp0 + col][kk + half2];        k0.y = K1[tp0 + col][kk + half2 + 1];
      k1.x = K1[tp0 + 16 + col][kk + half2];   k1.y = K1[tp0 + 16 + col][kk + half2 + 1];
      a00 = wmma4(q0, k0, a00);
      a01 = wmma4(q0, k1, a01);
      a10 = wmma4(q1, k0, a10);
      a11 = wmma4(q1, k1, a11);
    }
#pragma unroll
    for (int r = 0; r < 8; ++r) {
      int j0 = tj0 + r + half8;
      sout[(long)j0 * NN + tp0 + col]             = a00[r];
      sout[(long)j0 * NN + tp0 + 16 + col]        = a01[r];
      sout[(long)(j0 + 16) * NN + tp0 + col]      = a10[r];
      sout[(long)(j0 + 16) * NN + tp0 + 16 + col] = a11[r];
    }
  }
}

// ------- S2 per (b,h,j): s += Qt_j . K2t_j^T, scale, softmax over p -------
__global__ __launch_bounds__(256) void k_s2_softmax(const float* __restrict__ qkv,
                                                    float* __restrict__ s,
                                                    float scale) {
  int blk = blockIdx.x;
  int j = blk % NN, bh = blk / NN, h = bh % HH, b = bh / HH;
  __shared__ __align__(16) float Qt[NN][DD + PAD];
  __shared__ __align__(16) float K2[NN][DD + PAD];
  __shared__ __align__(16) float st[NN][NN + PAD];
  __shared__ float rmax[NN], rsum[NN];
  int tid = threadIdx.x;
#pragma unroll
  for (int r = 0; r < 3; ++r) {
    int idx4 = tid + r * 256;
    int i = idx4 >> 3, dq = (idx4 & 7) * 4;
    long rowoff = ((long)(b * NN + i) * NN + j) * FIVEC;
    *(float4*)&Qt[i][dq] = *(const float4*)&qkv[rowoff + h * DD + dq];
    *(float4*)&K2[i][dq] = *(const float4*)&qkv[rowoff + 2 * CC + h * DD + dq];
  }
  float* sbase = s + (long)bh * NN * NN * NN;
#pragma unroll
  for (int r = 0; r < 9; ++r) {                      // preload S1 tile (i,p): 2304 f4
    int idx4 = tid + r * 256;
    int i = idx4 / 24, p4 = (idx4 % 24) * 4;
    *(float4*)&st[i][p4] = *(const float4*)&sbase[((long)i * NN + j) * NN + p4];
  }
  __syncthreads();
  int lane = tid & 31, wv = tid >> 5;
  int row16 = lane & 15, half2 = (lane >> 4) << 1, half8 = (lane >> 4) * 8, col = lane & 15;
  for (int g = wv; g < 9; g += 8) {                  // 3x3 groups of 2x2 tiles
    int ti0 = (g / 3) * 32, tp0 = (g % 3) * 32;
    v8f a00 = {}, a01 = {}, a10 = {}, a11 = {};
#pragma unroll
    for (int kk = 0; kk < DD; kk += 4) {
      v2f q0, q1, k0, k1;
      q0.x = Qt[ti0 + row16][kk + half2];      q0.y = Qt[ti0 + row16][kk + half2 + 1];
      q1.x = Qt[ti0 + 16 + row16][kk + half2]; q1.y = Qt[ti0 + 16 + row16][kk + half2 + 1];
      k0.x = K2[tp0 + col][kk + half2];        k0.y = K2[tp0 + col][kk + half2 + 1];
      k1.x = K2[tp0 + 16 + col][kk + half2];   k1.y = K2[tp0 + 16 + col][kk + half2 + 1];
      a00 = wmma4(q0, k0, a00);
      a01 = wmma4(q0, k1, a01);
      a10 = wmma4(q1, k0, a10);
      a11 = wmma4(q1, k1, a11);
    }
#pragma unroll
    for (int r = 0; r < 8; ++r) {                    // unique ownership per lane
      int i0 = ti0 + r + half8;
      st[i0][tp0 + col]           = (st[i0][tp0 + col] + a00[r]) * scale;
      st[i0][tp0 + 16 + col]      = (st[i0][tp0 + 16 + col] + a01[r]) * scale;
      st[i0 + 16][tp0 + col]      = (st[i0 + 16][tp0 + col] + a10[r]) * scale;
      st[i0 + 16][tp0 + 16 + col] = (st[i0 + 16][tp0 + 16 + col] + a11[r]) * scale;
    }
  }
  __syncthreads();
  if (tid < NN) {
    float m = -1e30f;
    for (int p = 0; p < NN; ++p) m = fmaxf(m, st[tid][p]);
    rmax[tid] = m;
  }
  __syncthreads();
#pragma unroll
  for (int r = 0; r < 9; ++r) {
    int idx4 = tid + r * 256;
    int i = idx4 / 24, p4 = (idx4 % 24) * 4;
    float m = rmax[i];
    st[i][p4 + 0] = __expf(st[i][p4 + 0] - m);
    st[i][p4 + 1] = __expf(st[i][p4 + 1] - m);
    st[i][p4 + 2] = __expf(st[i][p4 + 2] - m);
    st[i][p4 + 3] = __expf(st[i][p4 + 3] - m);
  }
  __syncthreads();
  if (tid < NN) {
    float su = 0.f;
    for (int p = 0; p < NN; ++p) su += st[tid][p];
    rsum[tid] = 1.0f / su;
  }
  __syncthreads();
#pragma unroll
  for (int r = 0; r < 9; ++r) {
    int idx4 = tid + r * 256;
    int i = idx4 / 24, p4 = (idx4 % 24) * 4;
    float rs = rsum[i];
    float4 v;
    v.x = st[i][p4 + 0] * rs; v.y = st[i][p4 + 1] * rs;
    v.z = st[i][p4 + 2] * rs; v.w = st[i][p4 + 3] * rs;
    *(float4*)&sbase[((long)i * NN + j) * NN + p4] = v;
  }
}

// ---------------- O1 per (b,h,i): x = A_i . V1_i --------------------------
__global__ __launch_bounds__(256) void k_o1(const float* __restrict__ qkv,
                                            const float* __restrict__ s,
                                            float* __restrict__ x) {
  int blk = blockIdx.x;
  int i = blk % NN, bh = blk / NN, h = bh % HH, b = bh / HH;
  __shared__ __align__(16) float A[NN][NN + PAD];
  __shared__ __align__(16) float Vt[DD][NN + PAD];   // transposed: [d][p]
  int tid = threadIdx.x;
  const float* sbase = s + ((long)bh * NN + i) * NN * NN;   // contiguous [j][p]
#pragma unroll
  for (int r = 0; r < 9; ++r) {
    int idx4 = tid + r * 256;
    int j2 = idx4 / 24, p4 = (idx4 % 24) * 4;
    *(float4*)&A[j2][p4] = *(const float4*)&sbase[idx4 * 4];
  }
#pragma unroll
  for (int r = 0; r < 3; ++r) {
    int idx4 = tid + r * 256;
    int p = idx4 >> 3, dq = (idx4 & 7) * 4;
    float4 v = *(const float4*)&qkv[((long)(b * NN + i) * NN + p) * FIVEC + 3 * CC + h * DD + dq];
    Vt[dq + 0][p] = v.x; Vt[dq + 1][p] = v.y; Vt[dq + 2][p] = v.z; Vt[dq + 3][p] = v.w;
  }
  __syncthreads();
  int lane = tid & 31, wv = tid >> 5;
  int row16 = lane & 15, half2 = (lane >> 4) << 1, half8 = (lane >> 4) * 8, col = lane & 15;
  for (int t = wv; t < 12; t += 8) {                 // waves 0-3 reuse B (same td)
    int tj = t >> 1, td = t & 1;
    v8f acc = {};
#pragma unroll
    for (int kk = 0; kk < NN; kk += 4) {
      v2f a, bb;
      a.x  = A[tj * 16 + row16][kk + half2];   a.y  = A[tj * 16 + row16][kk + half2 + 1];
      bb.x = Vt[td * 16 + col][kk + half2];    bb.y = Vt[td * 16 + col][kk + half2 + 1];
      acc = wmma4(a, bb, acc);
    }
#pragma unroll
    for (int r = 0; r < 8; ++r) {
      int j = tj * 16 + r + half8, dd = td * 16 + col;
      x[((long)(b * NN + i) * NN + j) * CC + h * DD + dd] = acc[r];
    }
  }
}

// ---------------- O2 per (b,h,j): x += At_j . V2t_j -----------------------
__global__ __launch_bounds__(256) void k_o2(const float* __restrict__ qkv,
                                            const float* __restrict__ s,
                                            float* __restrict__ x) {
  int blk = blockIdx.x;
  int j = blk % NN, bh = blk / NN, h = bh % HH, b = bh / HH;
  __shared__ __align__(16) float A[NN][NN + PAD];
  __shared__ __align__(16) float Vt[DD][NN + PAD];   // transposed: [d][p]
  int tid = threadIdx.x;
  const float* sbase = s + (long)bh * NN * NN * NN;
#pragma unroll
  for (int r = 0; r < 9; ++r) {
    int idx4 = tid + r * 256;
    int i = idx4 / 24, p4 = (idx4 % 24) * 4;
    *(float4*)&A[i][p4] = *(const float4*)&sbase[((long)i * NN + j) * NN + p4];
  }
#pragma unroll
  for (int r = 0; r < 3; ++r) {
    int idx4 = tid + r * 256;
    int p = idx4 >> 3, dq = (idx4 & 7) * 4;
    float4 v = *(const float4*)&qkv[((long)(b * NN + p) * NN + j) * FIVEC + 4 * CC + h * DD + dq];
    Vt[dq + 0][p] = v.x; Vt[dq + 1][p] = v.y; Vt[dq + 2][p] = v.z; Vt[dq + 3][p] = v.w;
  }
  __syncthreads();
  int lane = tid & 31, wv = tid >> 5;
  int row16 = lane & 15, half2 = (lane >> 4) << 1, half8 = (lane >> 4) * 8, col = lane & 15;
  for (int t = wv; t < 12; t += 8) {
    int ti = t >> 1, td = t & 1;
    v8f acc = {};
#pragma unroll
    for (int kk = 0; kk < NN; kk += 4) {
      v2f a, bb;
      a.x  = A[ti * 16 + row16][kk + half2];   a.y  = A[ti * 16 + row16][kk + half2 + 1];
      bb.x = Vt[td * 16 + col][kk + half2];    bb.y = Vt[td * 16 + col][kk + half2 + 1];
      acc = wmma4(a, bb, acc);
    }
#pragma unroll
    for (int r = 0; r < 8; ++r) {
      int i = ti * 16 + r + half8, dd = td * 16 + col;
      long idx = ((long)(b * NN + i) * NN + j) * CC + h * DD + dd;
      x[idx] += acc[r];                              // disjoint across blocks
    }
  }
}

// ---------------- tail: mean over (i,j), then head linear -----------------
__global__ __launch_bounds__(256) void k_colmean1(const float* __restrict__ x,
                                                  float* __restrict__ mv1) {
  int seg = blockIdx.x, b = blockIdx.y, k = threadIdx.x;
  const float* p = x + ((long)b * NN * NN + (long)seg * 256) * CC + k;
  float su = 0.f;
  for (int r = 0; r < 256; ++r) su += p[(long)r * CC];
  mv1[((long)b * MSEG + seg) * CC + k] = su;
}

__global__ __launch_bounds__(256) void k_colmean2(const float* __restrict__ mv1,
                                                  float* __restrict__ mv) {
  int b = blockIdx.x, k = threadIdx.x;
  float su = 0.f;
  for (int sg = 0; sg < MSEG; ++sg) su += mv1[((long)b * MSEG + sg) * CC + k];
  mv[b * CC + k] = su / (float)(NN * NN);
}

__global__ __launch_bounds__(256) void k_head(const float* __restrict__ mv,
                                              const float* __restrict__ hw,
                                              const float* __restrict__ hb,
                                              float* __restrict__ out) {
  int b = blockIdx.x, co = threadIdx.x;
  __shared__ float m[CC];
  m[co] = mv[b * CC + co];
  __syncthreads();
  float su = hb[co];
  for (int k = 0; k < CC; ++k) su += m[k] * hw[(long)co * CC + k];
  out[b * CC + co] = su;
}

extern "C" void kernel_launch(void* const* d_in, const int* in_sizes, int n_in,
                              void* d_out, int out_size, void* d_ws, size_t ws_size,
                              hipStream_t stream) {
  const int*   adj    = (const int*)d_in[0];
  const float* emb_w  = (const float*)d_in[1];
  const float* qkv_w  = (const float*)d_in[2];
  const float* head_w = (const float*)d_in[3];
  const float* head_b = (const float*)d_in[4];
  float* out = (float*)d_out;

  float* xbuf  = (float*)d_ws;                 //  9,437,184 f
  float* qkvb  = xbuf + (long)MROWS * CC;      // 47,185,920 f
  float* sbuf  = qkvb + (long)MROWS * FIVEC;   // 28,311,552 f
  float* part  = sbuf + (long)BB * HH * NN * NN * NN;
  float* stats = part + 2 * REDBLK;
  float* mv    = stats + 2;
  float* mv1   = mv + BB * CC;                 // BB*MSEG*CC partials

  const float scale = 50.0f / sqrtf((float)DD);  // head_dim^-0.5 * 50

  k_embed<<<MROWS / 4, 256, 0, stream>>>(adj, emb_w, xbuf);

  for (int l = 0; l < DEPTH; ++l) {
    k_red1<<<REDBLK, 256, 0, stream>>>(xbuf, part);
    k_red2<<<1, 256, 0, stream>>>(part, stats);
    k_qkv<<<dim3(FIVEC / BN, MROWS / BM), 256, 0, stream>>>(
        xbuf, qkv_w + (long)l * FIVEC * CC, stats, qkvb);
    k_s1<<<BB * HH * NN, 256, 0, stream>>>(qkvb, sbuf);
    k_s2_softmax<<<BB * HH * NN, 256, 0, stream>>>(qkvb, sbuf, scale);
    k_o1<<<BB * HH * NN, 256, 0, stream>>>(qkvb, sbuf, xbuf);
    k_o2<<<BB * HH * NN, 256, 0, stream>>>(qkvb, sbuf, xbuf);
  }

  k_colmean1<<<dim3(MSEG, BB), 256, 0, stream>>>(xbuf, mv1);
  k_colmean2<<<BB, 256, 0, stream>>>(mv1, mv);
  k_head<<<BB, 256, 0, stream>>>(mv, head_w, head_b, out);
}